// GraphPooling_28587302322978
// MI455X (gfx1250) — compile-verified
//
#include <hip/hip_runtime.h>
#include <math.h>

// ---- problem constants (from reference) -------------------------------
namespace {
constexpr int   kNX   = 87;                 // floor(346/4)+1
constexpr int   kNY   = 66;                 // floor(260/4)+1
constexpr int   kB    = 16;
constexpr int   kGrid = kNX * kNY;          // 5742
constexpr int   kNSeg = kB * kGrid;         // 91872
constexpr int   kF    = 8;
constexpr float kInvS = 0.25f;              // 1/4.0, exact (power of two)
}

// Native CDNA5 float atomics (non-returning, device scope), SADDR form:
// uniform base in SGPR pair, 32-bit byte offset in one VGPR, per-element
// displacement in the instruction's immediate offset field.
template <int OFF>
__device__ __forceinline__ void amax_s(const float* base, unsigned voff, float v) {
  asm volatile("global_atomic_max_num_f32 %0, %1, %2 offset:%3 scope:SCOPE_DEV"
               :: "v"(voff), "v"(v), "s"(base), "i"(OFF)
               : "memory");
}
template <int OFF>
__device__ __forceinline__ void aadd_s(const float* base, unsigned voff, float v) {
  asm volatile("global_atomic_add_f32 %0, %1, %2 offset:%3 scope:SCOPE_DEV"
               :: "v"(voff), "v"(v), "s"(base), "i"(OFF)
               : "memory");
}

// ---------------------------------------------------------------------
// Scatter-reduce (defined first so the disasm snippet shows it).
// One thread per point:
//   8x global_atomic_max_num_f32  (segment_max over features)
//   4x global_atomic_add_f32      (segment_sum over [pos, 1])
// Accumulator tables (~4.4 MB) stay resident in the 192 MB L2.
// ---------------------------------------------------------------------
__global__ void gp_scatter_kernel(const float* __restrict__ x,
                                  const float* __restrict__ pos,
                                  const int*   __restrict__ batch,
                                  float* __restrict__ xmax,
                                  float* __restrict__ sums,
                                  int n) {
  const int i = blockIdx.x * blockDim.x + threadIdx.x;
  if (i >= n) return;

  const float px = pos[3 * (size_t)i + 0];
  const float py = pos[3 * (size_t)i + 1];
  const float pz = pos[3 * (size_t)i + 2];

  int cx = (int)floorf(px * kInvS);
  int cy = (int)floorf(py * kInvS);
  cx = min(max(cx, 0), kNX - 1);
  cy = min(max(cy, 0), kNY - 1);
  const int seg = batch[i] * kGrid + cy * kNX + cx;

  // 32 contiguous bytes per lane -> two global_load_b128, coalesced per wave.
  const float4* __restrict__ xr = (const float4*)(x + (size_t)i * kF);
  const float4 a0 = xr[0];
  const float4 a1 = xr[1];

  const unsigned offx = (unsigned)seg * (kF * 4);  // byte offset into xmax
  amax_s< 0>(xmax, offx, a0.x);
  amax_s< 4>(xmax, offx, a0.y);
  amax_s< 8>(xmax, offx, a0.z);
  amax_s<12>(xmax, offx, a0.w);
  amax_s<16>(xmax, offx, a1.x);
  amax_s<20>(xmax, offx, a1.y);
  amax_s<24>(xmax, offx, a1.z);
  amax_s<28>(xmax, offx, a1.w);

  const unsigned offs = (unsigned)seg * 16;        // byte offset into sums
  aadd_s< 0>(sums, offs, px);
  aadd_s< 4>(sums, offs, py);
  aadd_s< 8>(sums, offs, pz);
  aadd_s<12>(sums, offs, 1.0f);
}

// ---------------------------------------------------------------------
// Init: one b128 store per thread, no loops / no trip-count division.
//   xmax (= x_pool region of d_out): NSEG*8 floats -> -inf  (NSEG*2 float4)
//   sums (workspace):                NSEG*4 floats -> 0     (NSEG   float4)
// ---------------------------------------------------------------------
__global__ void gp_init_kernel(float4* __restrict__ xmax4,
                               float4* __restrict__ sums4) {
  const int i = blockIdx.x * blockDim.x + threadIdx.x;
  const float ninf = -__builtin_inff();
  if (i < kNSeg * 2) {
    xmax4[i] = make_float4(ninf, ninf, ninf, ninf);
  } else if (i < kNSeg * 3) {
    sums4[i - kNSeg * 2] = make_float4(0.0f, 0.0f, 0.0f, 0.0f);
  }
}

// ---------------------------------------------------------------------
// Finalize. One thread per segment.
//   x_pool    : mask empty segments to 0 (in place over the accumulator)
//   pos_pool  : sums / max(cnt,1), masked
//   batch_pool: seg / (NX*NY), written as raw int32
// d_out layout: [NSEG*8 f32 | NSEG*3 f32 | NSEG i32]
// ---------------------------------------------------------------------
__global__ void gp_finalize_kernel(float* __restrict__ out,
                                   const float* __restrict__ sums) {
  const int seg = blockIdx.x * blockDim.x + threadIdx.x;
  if (seg >= kNSeg) return;

  const float sx  = sums[(size_t)seg * 4 + 0];
  const float sy  = sums[(size_t)seg * 4 + 1];
  const float sz  = sums[(size_t)seg * 4 + 2];
  const float cnt = sums[(size_t)seg * 4 + 3];
  const bool  nz  = cnt > 0.0f;
  const float den = fmaxf(cnt, 1.0f);

  float* ox = out + (size_t)seg * kF;
#pragma unroll
  for (int f = 0; f < kF; ++f) {
    const float v = ox[f];
    ox[f] = nz ? v : 0.0f;
  }

  float* op = out + (size_t)kNSeg * kF + (size_t)seg * 3;
  op[0] = nz ? (sx / den) : 0.0f;
  op[1] = nz ? (sy / den) : 0.0f;
  op[2] = nz ? (sz / den) : 0.0f;

  int* ob = (int*)(out + (size_t)kNSeg * (kF + 3));
  ob[seg] = seg / kGrid;
}

// ---------------------------------------------------------------------
extern "C" void kernel_launch(void* const* d_in, const int* in_sizes, int n_in,
                              void* d_out, int out_size, void* d_ws, size_t ws_size,
                              hipStream_t stream) {
  const float* x     = (const float*)d_in[0];   // [N, 8] f32
  const float* pos   = (const float*)d_in[1];   // [N, 3] f32
  const int*   batch = (const int*)d_in[2];     // [N]    i32
  const int n = in_sizes[2];

  float* out  = (float*)d_out;
  float* xmax = out;               // reuse x_pool output region as max-accumulator
  float* sums = (float*)d_ws;      // NSEG*4 floats (~1.47 MB) of scratch

  const int init_threads = kNSeg * 3;  // float4 stores
  gp_init_kernel<<<(init_threads + 255) / 256, 256, 0, stream>>>(
      (float4*)xmax, (float4*)sums);

  const int blocks = (n + 255) / 256;
  gp_scatter_kernel<<<blocks, 256, 0, stream>>>(x, pos, batch, xmax, sums, n);

  gp_finalize_kernel<<<(kNSeg + 255) / 256, 256, 0, stream>>>(out, sums);
}